// A3TGCN_Temporal_35605278884395
// MI455X (gfx1250) — compile-verified
//
#include <hip/hip_runtime.h>
#include <hip/hip_bf16.h>

#define Nn 50000
#define Ee 800000
#define Ff 4
#define Tt 12
#define Cc 32
#define HORo 12
#define FT (Ff*Tt)   // 48 floats per node

typedef float v2f __attribute__((ext_vector_type(2)));
typedef float v8f __attribute__((ext_vector_type(8)));

__device__ __forceinline__ v8f wmma4(v2f a, v2f b, v8f c) {
  // D = A(16x4) * B(4x16) + C, all fp32 (exact vs reference fp32 math)
  return __builtin_amdgcn_wmma_f32_16x16x4_f32(false, a, false, b, (short)0, c, false, false);
}

// Branch-free hardware transcendentals (v_exp_f32 / v_rcp_f32 / v_tanh_f32)
__device__ __forceinline__ float fexp(float x) {
  return __builtin_amdgcn_exp2f(x * 1.4426950408889634f);
}
__device__ __forceinline__ float frcp(float x) { return __builtin_amdgcn_rcpf(x); }
// 1 - sigmoid(z) = 1/(1+e^z); exp->inf => rcp->0, exp->0 => 1. No branches needed.
__device__ __forceinline__ float one_minus_sigmoid(float z) {
  return frcp(1.f + fexp(z));
}
#if __has_builtin(__builtin_amdgcn_tanhf)
__device__ __forceinline__ float ftanh(float x) { return __builtin_amdgcn_tanhf(x); }
#else
__device__ __forceinline__ float ftanh(float x) {
  // tanh(x) = 1 - 2/(e^{2x}+1); saturates correctly at +/-inf without branches
  return 1.f - 2.f * frcp(fexp(2.f * x) + 1.f);
}
#endif

// ---------- prep: fold Wz@Wlz_top / Wh@Wlh_top, biases, softmax(attention) ----------
__global__ void prep_kernel(const float* __restrict__ Wz, const float* __restrict__ bz,
                            const float* __restrict__ Wlz, const float* __restrict__ blz,
                            const float* __restrict__ Wh, const float* __restrict__ bh,
                            const float* __restrict__ Wlh, const float* __restrict__ blh,
                            const float* __restrict__ att,
                            float* __restrict__ Wc, float* __restrict__ bc,
                            float* __restrict__ probs) {
  int tid = threadIdx.x;                // 256 threads = 4*64 elements of Wc
  int f = tid >> 6, c = tid & 63;
  float s = 0.f;
  if (c < Cc) {
    for (int j = 0; j < Cc; ++j) s += Wz[f*Cc + j] * Wlz[j*Cc + c];
  } else {
    int cc = c - Cc;
    for (int j = 0; j < Cc; ++j) s += Wh[f*Cc + j] * Wlh[j*Cc + cc];
  }
  Wc[f*64 + c] = s;
  if (tid < 64) {
    float bs;
    if (tid < Cc) {
      bs = blz[tid];
      for (int j = 0; j < Cc; ++j) bs += bz[j] * Wlz[j*Cc + tid];
    } else {
      int cc = tid - Cc;
      bs = blh[cc];
      for (int j = 0; j < Cc; ++j) bs += bh[j] * Wlh[j*Cc + cc];
    }
    bc[tid] = bs;
  }
  if (tid == 0) {
    float m = att[0];
    for (int t = 1; t < Tt; ++t) m = fmaxf(m, att[t]);
    float e[Tt], ss = 0.f;
    for (int t = 0; t < Tt; ++t) { e[t] = expf(att[t] - m); ss += e[t]; }
    for (int t = 0; t < Tt; ++t) probs[t] = e[t] / ss;
  }
}

// ---------- graph normalization ----------
__global__ void deg_init_kernel(float* __restrict__ deg) {
  int i = blockIdx.x * blockDim.x + threadIdx.x;
  if (i < Nn) deg[i] = 1.0f;
}
__global__ void deg_acc_kernel(const long long* __restrict__ ei,
                               const float* __restrict__ w, float* __restrict__ deg) {
  int e = blockIdx.x * blockDim.x + threadIdx.x;
  if (e < Ee) atomicAdd(&deg[(int)ei[Ee + e]], w[e]);
}
__global__ void dis_kernel(float* __restrict__ d) {
  int i = blockIdx.x * blockDim.x + threadIdx.x;
  if (i < Nn) d[i] = rsqrtf(d[i]);
}
__global__ void norm_kernel(const long long* __restrict__ ei, const float* __restrict__ w,
                            const float* __restrict__ dis, float* __restrict__ nv) {
  int e = blockIdx.x * blockDim.x + threadIdx.x;
  if (e < Ee) nv[e] = dis[(int)ei[e]] * w[e] * dis[(int)ei[Ee + e]];
}

// ---------- A(x): self term, then edge scatter (x first: 192B/node, L2-resident) ----------
__global__ void xagg_init_kernel(const float* __restrict__ x, const float* __restrict__ dis,
                                 float* __restrict__ Xagg) {
  int i = blockIdx.x * blockDim.x + threadIdx.x;
  if (i < Nn * FT) {
    int n = i / FT;
    float d = dis[n];
    Xagg[i] = d * d * x[i];
  }
}
// one thread per (edge, float4-chunk): b128 gather + 4 L2 atomics
__global__ void xagg_scatter_kernel(const long long* __restrict__ ei,
                                    const float* __restrict__ nv,
                                    const float4* __restrict__ x4,
                                    float* __restrict__ Xagg) {
  int i = blockIdx.x * blockDim.x + threadIdx.x;  // E*12 = 9.6M
  if (i < Ee * (FT / 4)) {
    int e = i / (FT / 4), q = i - e * (FT / 4);
    int r = (int)ei[e], cidx = (int)ei[Ee + e];
    float nm = nv[e];
    float4 v = x4[(size_t)r * (FT / 4) + q];
    float* dst = Xagg + (size_t)cidx * FT + q * 4;
    atomicAdd(dst + 0, nm * v.x);
    atomicAdd(dst + 1, nm * v.y);
    atomicAdd(dst + 2, nm * v.z);
    atomicAdd(dst + 3, nm * v.w);
  }
}

// ---------- gates + temporal accumulation + output GEMM, all via fp32 WMMA ----------
__global__ void __launch_bounds__(256) gate_out_kernel(
    const float* __restrict__ Xagg, const float* __restrict__ Wc,
    const float* __restrict__ bc, const float* __restrict__ probs,
    const float* __restrict__ Wout, const float* __restrict__ bout,
    float* __restrict__ out, int nTiles) {
  __shared__ float lds[8][16 * 33];           // per-wave 16x32 transpose tile (+pad)
  const int lane = threadIdx.x & 31;
  const int wv = threadIdx.x >> 5;            // 8 wave32 per block
  const int hf = lane >> 4;                   // lane half: 0/1
  const int lm = lane & 15;                   // M (A rows) or N (B/D cols)
  int wg = blockIdx.x * 8 + wv;
  bool active = wg < nTiles;
  int tile = active ? wg : (nTiles - 1);      // clamp: keep loads valid, EXEC all-ones

  // B fragments for the gate GEMM: 4 column tiles of Wc (4 x 64)
  v2f bg[4]; float cb[4];
#pragma unroll
  for (int g = 0; g < 4; ++g) {
    int col = g * 16 + lm;
    bg[g].x = Wc[(hf * 2 + 0) * 64 + col];
    bg[g].y = Wc[(hf * 2 + 1) * 64 + col];
    cb[g] = bc[col];
  }
  // B fragments for the output GEMM: Wout (32 x 12, N padded to 16)
  v2f bo[8];
#pragma unroll
  for (int ks = 0; ks < 8; ++ks) {
    int k0 = ks * 4 + hf * 2;
    bo[ks].x = (lm < HORo) ? Wout[(k0 + 0) * HORo + lm] : 0.f;
    bo[ks].y = (lm < HORo) ? Wout[(k0 + 1) * HORo + lm] : 0.f;
  }
  float ob = (lm < HORo) ? bout[lm] : 0.f;
  float p[Tt];
#pragma unroll
  for (int t = 0; t < Tt; ++t) p[t] = probs[t];

  const float* xb = Xagg + (size_t)tile * 16 * FT;
  v8f acc0 = {}, acc1 = {};
  for (int t = 0; t < Tt; ++t) {
    v2f a;                                    // A 16x4: lane%16=M, K=2*(lane/16)+vgpr
    a.x = xb[lm * FT + (hf * 2 + 0) * Tt + t];
    a.y = xb[lm * FT + (hf * 2 + 1) * Tt + t];
    v8f c0, c1, c2, c3;
#pragma unroll
    for (int v = 0; v < 8; ++v) { c0[v] = cb[0]; c1[v] = cb[1]; c2[v] = cb[2]; c3[v] = cb[3]; }
    v8f z0 = wmma4(a, bg[0], c0);             // z pre-act cols 0-15
    v8f z1 = wmma4(a, bg[1], c1);             // z cols 16-31
    v8f h0 = wmma4(a, bg[2], c2);             // h cols 0-15
    v8f h1 = wmma4(a, bg[3], c3);             // h cols 16-31
    float pt = p[t];
#pragma unroll
    for (int v = 0; v < 8; ++v) {
      acc0[v] += pt * one_minus_sigmoid(z0[v]) * ftanh(h0[v]);
      acc1[v] += pt * one_minus_sigmoid(z1[v]) * ftanh(h1[v]);
    }
  }
  // relu + transpose via LDS: D layout (lane=col, row=v+8*half) -> A layout (lane=row)
  float* L = lds[wv];
#pragma unroll
  for (int v = 0; v < 8; ++v) {
    int m = v + 8 * hf;
    L[m * 33 + lm]      = fmaxf(acc0[v], 0.f);
    L[m * 33 + 16 + lm] = fmaxf(acc1[v], 0.f);
  }
  __syncthreads();
  v8f o;
#pragma unroll
  for (int v = 0; v < 8; ++v) o[v] = ob;
#pragma unroll
  for (int ks = 0; ks < 8; ++ks) {            // K=32 as 8 chained K=4 WMMAs
    v2f a2;
    a2.x = L[lm * 33 + ks * 4 + hf * 2 + 0];
    a2.y = L[lm * 33 + ks * 4 + hf * 2 + 1];
    o = wmma4(a2, bo[ks], o);
  }
  if (active && lm < HORo) {
#pragma unroll
    for (int v = 0; v < 8; ++v) {
      int m = v + 8 * hf;
      out[((size_t)tile * 16 + m) * HORo + lm] = o[v];
    }
  }
}

extern "C" void kernel_launch(void* const* d_in, const int* in_sizes, int n_in,
                              void* d_out, int out_size, void* d_ws, size_t ws_size,
                              hipStream_t stream) {
  const float*     x    = (const float*)d_in[0];
  const long long* ei   = (const long long*)d_in[1];
  const float*     w    = (const float*)d_in[2];
  const float*     att  = (const float*)d_in[3];
  const float*     Wz   = (const float*)d_in[4];
  const float*     bz   = (const float*)d_in[5];
  const float*     Wlz  = (const float*)d_in[6];
  const float*     blz  = (const float*)d_in[7];
  // d_in[8..11] = Wr,br,Wlr,blr -> dead code in the reference (H0 == 0)
  const float*     Wh   = (const float*)d_in[12];
  const float*     bh   = (const float*)d_in[13];
  const float*     Wlh  = (const float*)d_in[14];
  const float*     blh  = (const float*)d_in[15];
  const float*     Wout = (const float*)d_in[16];
  const float*     bout = (const float*)d_in[17];
  float* out = (float*)d_out;

  float* wsf   = (float*)d_ws;
  float* dis   = wsf;                         // N   (deg, then rsqrt in place)
  float* nv    = wsf + 50176;                 // E
  float* Xagg  = wsf + 50176 + 800256;        // N*48
  float* Wc    = Xagg + 2400256;              // 4*64
  float* bc    = Wc + 256;                    // 64
  float* probs = bc + 64;                     // 12

  prep_kernel<<<1, 256, 0, stream>>>(Wz, bz, Wlz, blz, Wh, bh, Wlh, blh, att, Wc, bc, probs);
  deg_init_kernel<<<(Nn + 255) / 256, 256, 0, stream>>>(dis);
  deg_acc_kernel<<<(Ee + 255) / 256, 256, 0, stream>>>(ei, w, dis);
  dis_kernel<<<(Nn + 255) / 256, 256, 0, stream>>>(dis);
  norm_kernel<<<(Ee + 255) / 256, 256, 0, stream>>>(ei, w, dis, nv);
  xagg_init_kernel<<<(Nn * FT + 255) / 256, 256, 0, stream>>>(x, dis, Xagg);
  xagg_scatter_kernel<<<(Ee * (FT / 4) + 255) / 256, 256, 0, stream>>>(
      ei, nv, (const float4*)x, Xagg);

  int nTiles = Nn / 16;                       // 3125 exactly
  gate_out_kernel<<<(nTiles + 7) / 8, 256, 0, stream>>>(Xagg, Wc, bc, probs, Wout, bout, out, nTiles);
}